// GNN_73813307949751
// MI455X (gfx1250) — compile-verified
//
#include <hip/hip_runtime.h>
#include <hip/hip_bf16.h>

#define NHID   128
#define NOUT   64
#define NLAYER 4
#define BN_EPS 1e-5f

typedef __attribute__((ext_vector_type(2))) float v2f;
typedef __attribute__((ext_vector_type(8))) float v8f;

// ---------------------------------------------------------------------------
// WMMA fp32 strip GEMM: H[M x KOUT] = X[M x KIN] * W[KIN x KOUT] (+ bias)
// One wave (32 threads) computes a full 16-row x KOUT-col strip:
// NT = KOUT/16 accumulator tiles live in registers, the A fragment is loaded
// once per K-step and reused across all NT WMMAs (8x less A traffic).
// V_WMMA_F32_16X16X4_F32 layouts:
//   A 16x4 : lanes 0-15 M=lane, VGPR0=K0,VGPR1=K1; lanes 16-31: K2,K3
//   B 4x16 : VGPR0 = rows K0(lo-half)/K2(hi-half); VGPR1 = K1/K3, N=lane&15
//   C 16x16: VGPR v -> M = v + 8*(lane>>4), N = lane&15
// Grid: (M/16), block: 32, M multiple of 16 -> EXEC all ones (WMMA req).
// ---------------------------------------------------------------------------
template<int NT>
__global__ __launch_bounds__(32)
void gemm_strip_wmma(const float* __restrict__ X,
                     const float* __restrict__ W,
                     const float* __restrict__ bias,
                     float* __restrict__ H,
                     int KIN)
{
    constexpr int KOUT = NT * 16;
    const int lane = threadIdx.x & 31;
    const int lo   = lane & 15;
    const int hi   = lane >> 4;          // 0 or 1
    const int m0   = blockIdx.x << 4;

    const float* __restrict__ xrow = X + (size_t)(m0 + lo) * KIN;

    v8f acc[NT] = {};
    for (int k = 0; k < KIN; k += 4) {
        // A fragment: one 8B load, reused by all NT column tiles
        v2f a = *(const v2f*)(xrow + k + 2 * hi);
        const float* __restrict__ wr0 = W + (size_t)(k + 2 * hi) * KOUT + lo;
        const float* __restrict__ wr1 = wr0 + KOUT;
#pragma unroll
        for (int t = 0; t < NT; ++t) {
            v2f b;
            b.x = wr0[t * 16];
            b.y = wr1[t * 16];
            acc[t] = __builtin_amdgcn_wmma_f32_16x16x4_f32(
                         /*neg_a=*/false, a, /*neg_b=*/false, b,
                         /*c_mod=*/(short)0, acc[t],
                         /*reuse_a=*/false, /*reuse_b=*/false);
        }
    }

#pragma unroll
    for (int t = 0; t < NT; ++t) {
        const int col  = t * 16 + lo;
        const float ba = bias ? bias[col] : 0.0f;
#pragma unroll
        for (int v = 0; v < 8; ++v) {
            const int row = m0 + v + 8 * hi;
            H[(size_t)row * KOUT + col] = acc[t][v] + ba;
        }
    }
}

// ---------------------------------------------------------------------------
// Degree / normalization
// ---------------------------------------------------------------------------
__global__ void init_deg_kernel(float* __restrict__ deg, int n)
{
    int i = blockIdx.x * blockDim.x + threadIdx.x;
    if (i < n) deg[i] = 1.0f;            // self-loop contributes 1
}

__global__ void degree_kernel(const int* __restrict__ dst,
                              float* __restrict__ deg, int nE)
{
    int e = blockIdx.x * blockDim.x + threadIdx.x;
    if (e < nE) atomicAdd(&deg[dst[e]], 1.0f);
}

__global__ void dinv_kernel(float* __restrict__ deg, int n)
{
    int i = blockIdx.x * blockDim.x + threadIdx.x;
    if (i < n) deg[i] = rsqrtf(fmaxf(deg[i], 1.0f));   // in-place deg -> dinv
}

__global__ void copy_kernel(const float* __restrict__ src,
                            float* __restrict__ dst, int n)
{
    int i = blockIdx.x * blockDim.x + threadIdx.x;
    if (i < n) dst[i] = src[i];
}

// ---------------------------------------------------------------------------
// Fused edge + self-loop scatter:  agg[dst] += h[src] * dinv[src]*dinv[dst]
// One thread handles one (edge, 4-channel) slice: float4 gather, 4 atomics.
// Items [0,nE) are edges; items [nE, nE+nN) are self loops (norm = dinv^2).
// ---------------------------------------------------------------------------
__global__ void scatter_kernel(const float* __restrict__ h,
                               const int* __restrict__ src,
                               const int* __restrict__ dst,
                               const float* __restrict__ dinv,
                               float* __restrict__ agg,
                               int nE, int nN)
{
    const int tid   = blockIdx.x * blockDim.x + threadIdx.x;
    const int total = (nE + nN) * (NHID / 4);
    if (tid >= total) return;

    const int item = tid >> 5;               // NHID/4 == 32 slices per item
    const int c4   = (tid & 31) << 2;

    int s, d; float nrm;
    if (item < nE) {
        s = src[item]; d = dst[item];
        nrm = dinv[s] * dinv[d];
    } else {
        s = d = item - nE;
        const float di = dinv[s];
        nrm = di * di;
    }
    const float4 hv = *(const float4*)(h + (size_t)s * NHID + c4);
    float* out = agg + (size_t)d * NHID + c4;
    atomicAdd(out + 0, hv.x * nrm);
    atomicAdd(out + 1, hv.y * nrm);
    atomicAdd(out + 2, hv.z * nrm);
    atomicAdd(out + 3, hv.w * nrm);
}

// ---------------------------------------------------------------------------
// BatchNorm statistics: per-column sum and sum-of-squares.
// 128 threads/block, thread c owns column c -> fully coalesced row reads.
// ---------------------------------------------------------------------------
__global__ __launch_bounds__(NHID)
void bn_stats_kernel(const float* __restrict__ agg,
                     float* __restrict__ stats,   // [2*NHID]: sum, sumsq
                     int nN)
{
    const int c = threadIdx.x;
    float s = 0.f, s2 = 0.f;
    for (int r = blockIdx.x; r < nN; r += gridDim.x) {
        const float v = agg[(size_t)r * NHID + c];
        s  += v;
        s2 += v * v;
    }
    atomicAdd(&stats[c],        s);
    atomicAdd(&stats[NHID + c], s2);
}

__global__ __launch_bounds__(NHID)
void bn_finalize_kernel(const float* __restrict__ stats,
                        const float* __restrict__ gamma,
                        const float* __restrict__ beta,
                        float* __restrict__ ss,   // [2*NHID]: scale, shift
                        float invN)
{
    const int c = threadIdx.x;
    const float mean = stats[c] * invN;
    const float var  = stats[NHID + c] * invN - mean * mean;
    const float sc   = gamma[c] * rsqrtf(var + BN_EPS);
    ss[c]        = sc;
    ss[NHID + c] = beta[c] - mean * sc;
}

// x = relu(agg*scale + shift) + x   (residual, in place)
__global__ void bn_apply_kernel(const float* __restrict__ agg,
                                const float* __restrict__ ss,
                                float* __restrict__ xbuf,
                                int total)
{
    int i = blockIdx.x * blockDim.x + threadIdx.x;
    if (i >= total) return;
    const int c = i & (NHID - 1);
    float v = fmaf(agg[i], ss[c], ss[NHID + c]);
    v = fmaxf(v, 0.0f);
    xbuf[i] = v + xbuf[i];
}

// ---------------------------------------------------------------------------
extern "C" void kernel_launch(void* const* d_in, const int* in_sizes, int n_in,
                              void* d_out, int out_size, void* d_ws, size_t ws_size,
                              hipStream_t stream)
{
    const float* x_in    = (const float*)d_in[0];
    const int*   eidx    = (const int*)  d_in[1];
    const float* Ws      = (const float*)d_in[2];
    // d_in[3] (bs) is a no-op before BatchNorm -> skipped
    const float* gammas  = (const float*)d_in[4];
    const float* betas   = (const float*)d_in[5];
    const float* Wout    = (const float*)d_in[6];
    const float* bout    = (const float*)d_in[7];
    float*       out     = (float*)d_out;

    const int N  = in_sizes[0] / NHID;      // 50000
    const int E  = in_sizes[1] / 2;         // 800000
    const int NH = N * NHID;                // 6.4M floats

    const int* src = eidx;                  // edge_index[0]
    const int* dst = eidx + E;              // edge_index[1]

    // ---- workspace carve-up -------------------------------------------------
    float* ws    = (float*)d_ws;
    float* xbuf  = ws;                      // [N*NHID]
    float* hbuf  = xbuf + NH;               // [N*NHID]
    float* agg   = hbuf + NH;               // [N*NHID]
    float* dinv  = agg + NH;                // [N]   (deg then dinv in place)
    float* stats = dinv + ((N + 255) & ~255); // [2*NHID]
    float* ss    = stats + 2 * NHID;        // [2*NHID]

    const int T256 = 256;
    const int gN    = (N  + T256 - 1) / T256;
    const int gE    = (E  + T256 - 1) / T256;
    const int gNH   = (NH + T256 - 1) / T256;
    const int scat  = ((E + N) * (NHID / 4) + T256 - 1) / T256;

    // ---- degree normalization ----------------------------------------------
    init_deg_kernel<<<gN, T256, 0, stream>>>(dinv, N);
    degree_kernel  <<<gE, T256, 0, stream>>>(dst, dinv, E);
    dinv_kernel    <<<gN, T256, 0, stream>>>(dinv, N);

    // ---- x -> mutable buffer ------------------------------------------------
    copy_kernel<<<gNH, T256, 0, stream>>>(x_in, xbuf, NH);

    // ---- 4 GCN layers -------------------------------------------------------
    for (int l = 0; l < NLAYER; ++l) {
        gemm_strip_wmma<NHID / 16><<<N / 16, 32, 0, stream>>>(
            xbuf, Ws + (size_t)l * NHID * NHID, nullptr, hbuf, NHID);

        hipMemsetAsync(agg,   0, (size_t)NH * sizeof(float),       stream);
        hipMemsetAsync(stats, 0, (size_t)2 * NHID * sizeof(float), stream);

        scatter_kernel<<<scat, T256, 0, stream>>>(hbuf, src, dst, dinv, agg, E, N);

        bn_stats_kernel   <<<1024, NHID, 0, stream>>>(agg, stats, N);
        bn_finalize_kernel<<<1,    NHID, 0, stream>>>(stats,
                                                      gammas + l * NHID,
                                                      betas  + l * NHID,
                                                      ss, 1.0f / (float)N);
        bn_apply_kernel<<<gNH, T256, 0, stream>>>(agg, ss, xbuf, NH);
    }

    // ---- output head: out = x @ Wout + bout --------------------------------
    gemm_strip_wmma<NOUT / 16><<<N / 16, 32, 0, stream>>>(
        xbuf, Wout, bout, out, NHID);
}